// GCNModel_5068061409954
// MI455X (gfx1250) — compile-verified
//
#include <hip/hip_runtime.h>
#include <hip/hip_bf16.h>

#define NODES 100000
#define NEDGE 1600000
#define KDIM  128

typedef __attribute__((ext_vector_type(2))) float v2f;
typedef __attribute__((ext_vector_type(8))) float v8f;

// ------------------------- degree / symmetric norm -------------------------
__global__ void deg_init_k(float* __restrict__ deg, int n) {
  int i = blockIdx.x * blockDim.x + threadIdx.x;
  if (i < n) deg[i] = 1.0f;                       // self-loop contribution
}

__global__ void deg_edges_k(const long long* __restrict__ dst, float* __restrict__ deg, int e) {
  int i = blockIdx.x * blockDim.x + threadIdx.x;
  if (i < e) atomicAdd(&deg[dst[i]], 1.0f);
}

__global__ void deg_rsqrt_k(float* __restrict__ deg, int n) {
  int i = blockIdx.x * blockDim.x + threadIdx.x;
  if (i < n) deg[i] = rsqrtf(deg[i]);             // deg >= 1 always (self loops)
}

// ------------------------- GEMM: H = (relu?)(A) @ W -------------------------
// A: [n, 128] row-major; W: [128, F] row-major; Hout: [n, F]
// One 16x16 output tile per wave via V_WMMA_F32_16X16X4_F32, K unrolled 32x.
template<int F, bool RELU_A>
__global__ void gemm_wmma_k(const float* __restrict__ A,
                            const float* __restrict__ W,
                            float* __restrict__ Hout) {
  const int lane  = threadIdx.x & 31;
  const int wave  = threadIdx.x >> 5;             // one 16-col tile per wave
  const int mBase = blockIdx.x << 4;
  const int nBase = wave << 4;
  const int hi    = lane >> 4;                    // lane half: 0 or 1
  const int kb    = hi << 1;                      // K sub-offset: 0 or 2
  const int mRow  = mBase + (lane & 15);
  const int col   = nBase + (lane & 15);

  const float* __restrict__ arow = A + (size_t)mRow * KDIM;
  v8f acc = {};
#pragma unroll
  for (int k = 0; k < KDIM; k += 4) {
    // A fragment 16x4: lane = M row; VGPR0/1 hold K = kb, kb+1
    float a0 = arow[k + kb];
    float a1 = arow[k + kb + 1];
    if (RELU_A) { a0 = fmaxf(a0, 0.0f); a1 = fmaxf(a1, 0.0f); }
    v2f a; a.x = a0; a.y = a1;
    // B fragment 4x16: lane = N col; VGPR0/1 hold K = kb, kb+1
    v2f b; b.x = W[(size_t)(k + kb) * F + col];
           b.y = W[(size_t)(k + kb + 1) * F + col];
    acc = __builtin_amdgcn_wmma_f32_16x16x4_f32(false, a, false, b,
                                                (short)0, acc, false, false);
  }
  // C/D layout: VGPR r -> row r (lanes 0-15) / row r+8 (lanes 16-31)
  float* __restrict__ orow = Hout + (size_t)(mBase + (hi << 3)) * F + col;
#pragma unroll
  for (int r = 0; r < 8; ++r)
    orow[(size_t)r * F] = acc[r];
}

// ------------------------- aggregation -------------------------
// out[i][f] = bias[f] + h[i][f] * dinv[i]^2   (self-loop term; also inits buffer)
template<int F>
__global__ void agg_init_k(const float* __restrict__ h, const float* __restrict__ dinv,
                           const float* __restrict__ bias, float* __restrict__ out, int n) {
  long long idx = (long long)blockIdx.x * blockDim.x + threadIdx.x;
  if (idx < (long long)n * F) {
    int i = (int)(idx / F);
    int f = (int)(idx & (F - 1));
    float di = dinv[i];
    out[idx] = bias[f] + h[idx] * di * di;
  }
}

// One wave per edge: float4 per lane; 512B coalesced gather + fp32 atomic scatter.
template<int F>
__global__ void agg_edges_k(const float* __restrict__ h, const float* __restrict__ dinv,
                            const long long* __restrict__ src, const long long* __restrict__ dst,
                            float* __restrict__ out, int e) {
  int eid = blockIdx.x * (blockDim.x >> 5) + (threadIdx.x >> 5);
  if (eid >= e) return;
  const int lane = threadIdx.x & 31;
  const long long s = src[eid];
  const long long d = dst[eid];
  const float nrm = dinv[(int)s] * dinv[(int)d];
  const int f = lane << 2;
  if (f < F) {
    const float4 hv = *(const float4*)(h + (size_t)s * F + f);
    float* o = out + (size_t)d * F + f;
    atomicAdd(o + 0, hv.x * nrm);
    atomicAdd(o + 1, hv.y * nrm);
    atomicAdd(o + 2, hv.z * nrm);
    atomicAdd(o + 3, hv.w * nrm);
  }
}

// ------------------------- final mean over nodes -------------------------
__global__ void out_zero_k(float* __restrict__ out) { out[threadIdx.x] = 0.0f; } // <<<1,64>>>

__global__ void mean_reduce_k(const float* __restrict__ g, float* __restrict__ out, int n) {
  const int t   = threadIdx.x;
  const int f   = t & 63;
  const int grp = t >> 6;                          // 4 row-groups per block
  float acc = 0.0f;
  for (int r = blockIdx.x * 4 + grp; r < n; r += gridDim.x * 4)
    acc += g[(size_t)r * 64 + f];
  __shared__ float sm[256];
  sm[t] = acc;
  __syncthreads();
  if (grp == 0) {
    float v = sm[f] + sm[f + 64] + sm[f + 128] + sm[f + 192];
    atomicAdd(&out[f], v * (1.0f / (float)n));
  }
}

// ------------------------- launch -------------------------
extern "C" void kernel_launch(void* const* d_in, const int* in_sizes, int n_in,
                              void* d_out, int out_size, void* d_ws, size_t ws_size,
                              hipStream_t stream) {
  const float*     x    = (const float*)d_in[0];
  const float*     W1   = (const float*)d_in[1];
  const float*     b1   = (const float*)d_in[2];
  const float*     W2   = (const float*)d_in[3];
  const float*     b2   = (const float*)d_in[4];
  const float*     W3   = (const float*)d_in[5];
  const float*     b3   = (const float*)d_in[6];
  const long long* ei   = (const long long*)d_in[7];   // int64 edge_index [2, E]
  const long long* srcI = ei;
  const long long* dstI = ei + NEDGE;
  float*           outp = (float*)d_out;

  // Workspace: dinv [N] | bufH [N*128] | bufA [N*128]   (~103 MB)
  float* dinv = (float*)d_ws;
  float* bufH = dinv + ((NODES + 255) & ~255);
  float* bufA = bufH + (size_t)NODES * 128;

  const int TPB  = 256;
  const int nBlk = (NODES + TPB - 1) / TPB;
  const int eBlk = (NEDGE + TPB - 1) / TPB;
  const int MT   = NODES / 16;                 // 6250 row tiles (exact)
  const int eWav = (NEDGE + 7) / 8;            // 8 edge-waves per 256-thread block
  const int a128 = (int)(((long long)NODES * 128 + TPB - 1) / TPB);
  const int a64  = (int)(((long long)NODES * 64 + TPB - 1) / TPB);

  // symmetric normalization D^{-1/2}
  deg_init_k <<<nBlk, TPB, 0, stream>>>(dinv, NODES);
  deg_edges_k<<<eBlk, TPB, 0, stream>>>(dstI, dinv, NEDGE);
  deg_rsqrt_k<<<nBlk, TPB, 0, stream>>>(dinv, NODES);

  // layer 1: h = x@W1 ; agg + b1 (relu deferred into next GEMM's A-read)
  gemm_wmma_k<128, false><<<MT, 256, 0, stream>>>(x, W1, bufH);
  agg_init_k<128> <<<a128, TPB, 0, stream>>>(bufH, dinv, b1, bufA, NODES);
  agg_edges_k<128><<<eWav, 256, 0, stream>>>(bufH, dinv, srcI, dstI, bufA, NEDGE);

  // layer 2
  gemm_wmma_k<128, true><<<MT, 256, 0, stream>>>(bufA, W2, bufH);
  agg_init_k<128> <<<a128, TPB, 0, stream>>>(bufH, dinv, b2, bufA, NODES);
  agg_edges_k<128><<<eWav, 256, 0, stream>>>(bufH, dinv, srcI, dstI, bufA, NEDGE);

  // layer 3 (F = 64; 4 waves/block)
  gemm_wmma_k<64, true><<<MT, 128, 0, stream>>>(bufA, W3, bufH);
  agg_init_k<64> <<<a64, TPB, 0, stream>>>(bufH, dinv, b3, bufA, NODES);
  agg_edges_k<64><<<eWav, 256, 0, stream>>>(bufH, dinv, srcI, dstI, bufA, NEDGE);

  // mean over nodes -> [1, 64]
  out_zero_k  <<<1, 64, 0, stream>>>(outp);
  mean_reduce_k<<<512, 256, 0, stream>>>(bufA, outp, NODES);
}